// TensorizedSpectralConv_6743098655140
// MI455X (gfx1250) — compile-verified
//
#include <hip/hip_runtime.h>
#include <math.h>

// ---------------------------------------------------------------------------
// TensorizedSpectralConv on MI455X (gfx1250): truncated-mode FFT == dense DFT
// GEMMs -> V_WMMA_F32_16X16X4_F32 throughout. wave32 tiles: each wave owns
// 16x16 C tiles; complex GEMM = 4 real WMMA accumulators combined at store.
// Register blocking: waves share B fragments across m-tiles (fwd stages) and
// keep the whole A operand resident across the n-sweep (inverse stages).
// ---------------------------------------------------------------------------

typedef float v2f __attribute__((ext_vector_type(2)));
typedef float v8f __attribute__((ext_vector_type(8)));

static __device__ __forceinline__ v8f wmma4(v2f a, v2f b, v8f c) {
  // D = A(16x4 f32) * B(4x16 f32) + C(16x16 f32)
  return __builtin_amdgcn_wmma_f32_16x16x4_f32(false, a, false, b, (short)0, c,
                                               false, false);
}

// A fragment (16x4 f32): lanes 0-15 -> rows M=0..15 holding K=k0,k0+1;
// lanes 16-31 -> same rows holding K=k0+2,k0+3 (ISA 32-bit A layout).
static __device__ __forceinline__ v2f ldA(const float* __restrict__ A, int lda,
                                          int m0, int k0, int lane) {
  int row = m0 + (lane & 15);
  int kb  = k0 + ((lane >> 4) << 1);
  v2f a;
  a.x = A[(size_t)row * lda + kb];
  a.y = A[(size_t)row * lda + kb + 1];
  return a;
}

// B fragment (4x16 f32): lanes 0-15 carry rows K=k0,k0+1; lanes 16-31 rows
// K=k0+2,k0+3; column = n0 + (lane&15).
static __device__ __forceinline__ v2f ldB(const float* __restrict__ Bp, int ldb,
                                          int k0, int n0, int lane) {
  int col = n0 + (lane & 15);
  int kb  = k0 + ((lane >> 4) << 1);
  v2f b;
  b.x = Bp[(size_t)kb * ldb + col];
  b.y = Bp[(size_t)(kb + 1) * ldb + col];
  return b;
}

// C/D (16x16 f32, 8 VGPRs): lane<16 -> rows m0+0..7, lane>=16 -> rows m0+8..15.
static __device__ __forceinline__ void stC(float* __restrict__ C, int ldc,
                                           int m0, int n0, int lane, v8f c,
                                           int mlim) {
  int col = n0 + (lane & 15);
  int rb  = m0 + ((lane >> 4) << 3);
#pragma unroll
  for (int v = 0; v < 8; ++v) {
    int row = rb + v;
    if (row < mlim) C[(size_t)row * ldc + col] = c[v];
  }
}

// ---------------- problem constants ----------------
#define NB 32
#define NH 128
#define NW 128
#define NC 128
#define MH 32   // kept modes axis 0
#define MW 17   // rfft half-spectrum modes axis 1
#define MWP 20  // MW padded to multiple of 4 for WMMA K dim

// ---------------- small precompute kernels ----------------

__global__ void precomp_kernel(float* FwR, float* FwI, float* FhR, float* FhI,
                               float* GhR, float* GhI, float* GwCt, float* GwSt) {
  int id = blockIdx.x * blockDim.x + threadIdx.x;
  if (id >= 32 * 128) return;
  int m = id >> 7, y = id & 127;
  const float inv128 = 1.0f / 128.0f;
  const float w0 = 6.283185307179586f / 128.0f;
  // Fw: e^{-2pi i w x /128}/128, rows >=17 zero (M padding)
  if (m < MW) {
    int k = (m * y) & 127;
    float t = w0 * (float)k;
    FwR[id] = cosf(t) * inv128;
    FwI[id] = -sinf(t) * inv128;
  } else {
    FwR[id] = 0.f;
    FwI[id] = 0.f;
  }
  // Fh: kept modes m<16 -> freq m, m>=16 -> freq 96+m (negative quadrant)
  int f = (m < 16) ? m : (96 + m);
  {
    int k = (f * y) & 127;
    float t = w0 * (float)k;
    FhR[id] = cosf(t) * inv128;
    FhI[id] = -sinf(t) * inv128;
    // Gh[y][m] = e^{+2pi i f y/128} (inverse, norm='forward' -> no scale)
    GhR[y * 32 + m] = cosf(t);
    GhI[y * 32 + m] = sinf(t);
  }
  // GwCt[x][w] = a_w cos, GwSt[x][w] = -a_w sin ; cols 17..19 zero pad
  if (m < MWP) {
    int idx = y * MWP + m;
    if (m < MW) {
      int k = (m * y) & 127;
      float t = w0 * (float)k;
      float a = (m == 0) ? 1.f : 2.f;
      GwCt[idx] = a * cosf(t);
      GwSt[idx] = -a * sinf(t);
    } else {
      GwCt[idx] = 0.f;
      GwSt[idx] = 0.f;
    }
  }
}

// S[h][d][i][o] = sum_r kout[h,o,d,r] * kin[h,i,d,r]  (complex)
__global__ void buildS_kernel(const float* __restrict__ kout,
                              const float* __restrict__ kin, float* __restrict__ Sr,
                              float* __restrict__ Si, int total) {
  int id = blockIdx.x * 256 + threadIdx.x;
  if (id >= total) return;
  int o = id & 127;
  int i = (id >> 7) & 127;
  int d = (id >> 14) & 1;
  int h = id >> 15;
  const float* pi = kin + (size_t)(((h * 128 + i) * 2 + d) * 64) * 2;
  const float* po = kout + (size_t)(((h * 128 + o) * 2 + d) * 64) * 2;
  float sr = 0.f, si = 0.f;
#pragma unroll 8
  for (int r = 0; r < 64; ++r) {
    float ar = pi[2 * r], ai = pi[2 * r + 1];
    float br = po[2 * r], bi = po[2 * r + 1];
    sr += ar * br - ai * bi;
    si += ar * bi + ai * br;
  }
  Sr[id] = sr;
  Si[id] = si;
}

__global__ void zero_kernel(float* p, int n) {
  int i = blockIdx.x * 256 + threadIdx.x;
  if (i < n) p[i] = 0.f;
}

// ---------------- stage 1: W-axis forward DFT (real input) ----------------
// T1[b,y,w,c] = sum_x Fw[w,x] * X[b,y,x,c]
// 8 waves = 8 n-columns; each wave does both m-tiles jointly, sharing B loads.
__global__ void fwdW_kernel(const float* __restrict__ X, const float* __restrict__ FwR,
                            const float* __restrict__ FwI, float* __restrict__ T1r,
                            float* __restrict__ T1i) {
  int slice = blockIdx.x;  // b*128 + y, 4096 slices
  int wave = threadIdx.x >> 5, lane = threadIdx.x & 31;
  const float* Bx = X + (size_t)slice * (NW * NC);  // [x][c]
  float* outR = T1r + (size_t)slice * (MW * NC);
  float* outI = T1i + (size_t)slice * (MW * NC);
  int n0 = wave << 4;
  v8f cr0 = {}, ci0 = {}, cr1 = {}, ci1 = {};
  for (int k0 = 0; k0 < 128; k0 += 4) {
    v2f b   = ldB(Bx, NC, k0, n0, lane);  // shared across both m-tiles
    v2f ar0 = ldA(FwR, 128, 0, k0, lane);
    v2f ai0 = ldA(FwI, 128, 0, k0, lane);
    v2f ar1 = ldA(FwR, 128, 16, k0, lane);
    v2f ai1 = ldA(FwI, 128, 16, k0, lane);
    cr0 = wmma4(ar0, b, cr0);
    ci0 = wmma4(ai0, b, ci0);
    cr1 = wmma4(ar1, b, cr1);
    ci1 = wmma4(ai1, b, ci1);
  }
  stC(outR, NC, 0, n0, lane, cr0, MW);
  stC(outI, NC, 0, n0, lane, ci0, MW);
  stC(outR, NC, 16, n0, lane, cr1, MW);  // only row 16 (w=16) lands
  stC(outI, NC, 16, n0, lane, ci1, MW);
}

// ---------------- stage 2: H-axis forward DFT (complex) ----------------
// Xf[b,h,w,c] = sum_y Fh[h,y] * T1[b,y,w,c]
// 8 waves = 8 n-columns; both m-tiles per wave share the (complex) B loads.
__global__ void fwdH_kernel(const float* __restrict__ T1r, const float* __restrict__ T1i,
                            const float* __restrict__ FhR, const float* __restrict__ FhI,
                            float* __restrict__ Xfr, float* __restrict__ Xfi) {
  int slice = blockIdx.x;  // 544 = b*17 + w
  int b = slice / MW, w = slice % MW;
  int wave = threadIdx.x >> 5, lane = threadIdx.x & 31;
  size_t bbase = (size_t)b * NH * MW * NC + (size_t)w * NC;  // (y,c) stride MW*NC
  size_t obase = (size_t)b * MH * MW * NC + (size_t)w * NC;
  const int ldb = MW * NC;  // 2176
  int n0 = wave << 4;
  v8f crr0 = {}, cii0 = {}, cri0 = {}, cir0 = {};
  v8f crr1 = {}, cii1 = {}, cri1 = {}, cir1 = {};
  for (int k0 = 0; k0 < 128; k0 += 4) {
    v2f br = ldB(T1r + bbase, ldb, k0, n0, lane);  // shared across m-tiles
    v2f bi = ldB(T1i + bbase, ldb, k0, n0, lane);
    v2f ar0 = ldA(FhR, 128, 0, k0, lane);
    v2f ai0 = ldA(FhI, 128, 0, k0, lane);
    v2f ar1 = ldA(FhR, 128, 16, k0, lane);
    v2f ai1 = ldA(FhI, 128, 16, k0, lane);
    crr0 = wmma4(ar0, br, crr0);
    cii0 = wmma4(ai0, bi, cii0);
    cri0 = wmma4(ar0, bi, cri0);
    cir0 = wmma4(ai0, br, cir0);
    crr1 = wmma4(ar1, br, crr1);
    cii1 = wmma4(ai1, bi, cii1);
    cri1 = wmma4(ar1, bi, cri1);
    cir1 = wmma4(ai1, br, cir1);
  }
  v8f outr0 = crr0 - cii0, outi0 = cri0 + cir0;
  v8f outr1 = crr1 - cii1, outi1 = cri1 + cir1;
  stC(Xfr + obase, ldb, 0, n0, lane, outr0, MH);
  stC(Xfi + obase, ldb, 0, n0, lane, outi0, MH);
  stC(Xfr + obase, ldb, 16, n0, lane, outr1, MH);
  stC(Xfi + obase, ldb, 16, n0, lane, outi1, MH);
}

// ---------------- stage 3: spectral multiply with fused low-rank K ---------
// Y[b,h,w,o] = sum_i K[h,w,o,i] * Xf[b,h,w,i],
// K[o,i] = sum_d S0[h][d][i][o] * S1[w][d][i][o]  (complex, built in LDS)
__global__ void spectral_kernel(const float* __restrict__ Xfr, const float* __restrict__ Xfi,
                                const float* __restrict__ S0r, const float* __restrict__ S0i,
                                const float* __restrict__ S1r, const float* __restrict__ S1i,
                                float* __restrict__ Yr, float* __restrict__ Yi) {
  __shared__ float ktr[128 * 16];
  __shared__ float kti[128 * 16];
  int slice = blockIdx.x;  // 544 = h*17 + w
  int h = slice / MW, w = slice % MW;
  int n0g = blockIdx.y << 4;  // o-column of 16
  int tid = threadIdx.x;      // blockDim = 64 (2 waves)

  // build K^T tile [i=0..127][o=n0g..n0g+15] (complex) in LDS
  for (int idx = tid; idx < 128 * 16; idx += 64) {
    int i = idx >> 4;
    int o = n0g + (idx & 15);
    float br = 0.f, bi = 0.f;
#pragma unroll
    for (int d = 0; d < 2; ++d) {
      size_t p0 = ((size_t)(h * 2 + d) * 128 + i) * 128 + o;
      size_t p1 = ((size_t)(w * 2 + d) * 128 + i) * 128 + o;
      float s0r = S0r[p0], s0i = S0i[p0];
      float s1r = S1r[p1], s1i = S1i[p1];
      br += s0r * s1r - s0i * s1i;
      bi += s0r * s1i + s0i * s1r;
    }
    ktr[idx] = br;
    kti[idx] = bi;
  }
  __syncthreads();

  int wave = tid >> 5, lane = tid & 31;
  int m0 = wave << 4;  // b rows, M = 32
  size_t abase = ((size_t)h * MW + w) * NC;      // + b*lda, lda = MH*MW*NC
  const int lda = MH * MW * NC;                  // 69632
  v8f crr = {}, cii = {}, cri = {}, cir = {};
  for (int k0 = 0; k0 < 128; k0 += 4) {
    v2f ar = ldA(Xfr + abase, lda, m0, k0, lane);
    v2f ai = ldA(Xfi + abase, lda, m0, k0, lane);
    v2f br = ldB(ktr, 16, k0, 0, lane);
    v2f bi = ldB(kti, 16, k0, 0, lane);
    crr = wmma4(ar, br, crr);
    cii = wmma4(ai, bi, cii);
    cri = wmma4(ar, bi, cri);
    cir = wmma4(ai, br, cir);
  }
  v8f outr = crr - cii;
  v8f outi = cri + cir;
  size_t ybase = ((size_t)h * MWP + w) * NC;  // + b*ldc, ldc = MH*MWP*NC
  const int ldc = MH * MWP * NC;              // 81920
  stC(Yr + ybase, ldc, m0, n0g, lane, outr, MH);
  stC(Yi + ybase, ldc, m0, n0g, lane, outi, MH);
}

// ---------------- stage 4: H-axis inverse DFT (complex) ----------------
// T2[b,y,w,o] = sum_m Gh[y,m] * Y[b,m,w,o]
// 8 waves = 8 m-tiles; whole A operand (K=32) hoisted into registers,
// inner n-sweep issues only B loads.
__global__ void invH_kernel(const float* __restrict__ Yr, const float* __restrict__ Yi,
                            const float* __restrict__ GhR, const float* __restrict__ GhI,
                            float* __restrict__ T2r, float* __restrict__ T2i) {
  int slice = blockIdx.x;  // 640 = b*20 + w
  int b = slice / MWP, w = slice % MWP;
  int wave = threadIdx.x >> 5, lane = threadIdx.x & 31;
  size_t bbase = (size_t)b * MH * MWP * NC + (size_t)w * NC;  // (m,o), ldb = MWP*NC
  size_t obase = (size_t)b * NH * MWP * NC + (size_t)w * NC;  // (y,o), ldc = MWP*NC
  const int ldb = MWP * NC;  // 2560
  int m0 = wave << 4;        // y tile for this wave

  v2f ahr[8], ahi[8];
#pragma unroll
  for (int kk = 0; kk < 8; ++kk) {
    ahr[kk] = ldA(GhR, 32, m0, kk * 4, lane);
    ahi[kk] = ldA(GhI, 32, m0, kk * 4, lane);
  }

  for (int nt = 0; nt < 8; ++nt) {
    int n0 = nt << 4;
    v8f crr = {}, cii = {}, cri = {}, cir = {};
#pragma unroll
    for (int kk = 0; kk < 8; ++kk) {
      v2f br = ldB(Yr + bbase, ldb, kk * 4, n0, lane);
      v2f bi = ldB(Yi + bbase, ldb, kk * 4, n0, lane);
      crr = wmma4(ahr[kk], br, crr);
      cii = wmma4(ahi[kk], bi, cii);
      cri = wmma4(ahr[kk], bi, cri);
      cir = wmma4(ahi[kk], br, cir);
    }
    v8f outr = crr - cii;
    v8f outi = cri + cir;
    stC(T2r + obase, ldb, m0, n0, lane, outr, NH);
    stC(T2i + obase, ldb, m0, n0, lane, outi, NH);
  }
}

// ---------------- stage 5: W-axis inverse (real output) ----------------
// out[b,y,x,o] = sum_w GwCt[x,w]*T2r[w,o] + GwSt[x,w]*T2i[w,o]
// 8 waves = 8 m-tiles; A (K=20, both matrices) hoisted into registers.
__global__ void invW_kernel(const float* __restrict__ T2r, const float* __restrict__ T2i,
                            const float* __restrict__ GwCt, const float* __restrict__ GwSt,
                            float* __restrict__ out) {
  int slice = blockIdx.x;  // 4096 = b*128 + y
  int wave = threadIdx.x >> 5, lane = threadIdx.x & 31;
  size_t bbase = (size_t)slice * (MWP * NC);  // (w,o)
  size_t obase = (size_t)slice * (NW * NC);   // (x,o)
  int m0 = wave << 4;                         // x tile for this wave

  v2f ac[5], as[5];
#pragma unroll
  for (int kk = 0; kk < 5; ++kk) {
    ac[kk] = ldA(GwCt, MWP, m0, kk * 4, lane);
    as[kk] = ldA(GwSt, MWP, m0, kk * 4, lane);
  }

  for (int nt = 0; nt < 8; ++nt) {
    int n0 = nt << 4;
    v8f c = {};
#pragma unroll
    for (int kk = 0; kk < 5; ++kk) {
      v2f b1 = ldB(T2r + bbase, NC, kk * 4, n0, lane);
      c = wmma4(ac[kk], b1, c);
      v2f b2 = ldB(T2i + bbase, NC, kk * 4, n0, lane);
      c = wmma4(as[kk], b2, c);
    }
    stC(out + obase, NC, m0, n0, lane, c, NW);
  }
}

// ---------------- host side ----------------

extern "C" void kernel_launch(void* const* d_in, const int* in_sizes, int n_in,
                              void* d_out, int out_size, void* d_ws, size_t ws_size,
                              hipStream_t stream) {
  (void)in_sizes; (void)n_in; (void)out_size; (void)ws_size;
  const float* X     = (const float*)d_in[0];
  const float* kout0 = (const float*)d_in[1];
  const float* kin0  = (const float*)d_in[2];
  const float* kout1 = (const float*)d_in[3];
  const float* kin1  = (const float*)d_in[4];
  float* out = (float*)d_out;
  float* ws  = (float*)d_ws;

  // workspace layout (floats); T1 and T2 alias (disjoint lifetimes)
  const size_t SH_PLANE = (size_t)NB * NH * MWP * NC;  // 10,485,760 (T2 plane; T1 fits)
  const size_t XF_PLANE = (size_t)NB * MH * MW * NC;   // 2,228,224
  const size_t Y_PLANE  = (size_t)NB * MH * MWP * NC;  // 2,621,440
  const size_t S0_PLANE = (size_t)MH * 2 * 128 * 128;  // 1,048,576
  const size_t S1_PLANE = (size_t)MW * 2 * 128 * 128;  // 557,056

  size_t off = 0;
  float* T1r = ws + off; off += SH_PLANE;   float* T2r = T1r;
  float* T1i = ws + off; off += SH_PLANE;   float* T2i = T1i;
  float* Xfr = ws + off; off += XF_PLANE;
  float* Xfi = ws + off; off += XF_PLANE;
  float* Yr  = ws + off; off += Y_PLANE;
  float* Yi  = ws + off; off += Y_PLANE;
  float* S0r = ws + off; off += S0_PLANE;
  float* S0i = ws + off; off += S0_PLANE;
  float* S1r = ws + off; off += S1_PLANE;
  float* S1i = ws + off; off += S1_PLANE;
  float* FwR = ws + off; off += 32 * 128;
  float* FwI = ws + off; off += 32 * 128;
  float* FhR = ws + off; off += 32 * 128;
  float* FhI = ws + off; off += 32 * 128;
  float* GhR = ws + off; off += 128 * 32;
  float* GhI = ws + off; off += 128 * 32;
  float* GwC = ws + off; off += 128 * MWP;
  float* GwS = ws + off; off += 128 * MWP;

  // DFT factor matrices
  precomp_kernel<<<16, 256, 0, stream>>>(FwR, FwI, FhR, FhI, GhR, GhI, GwC, GwS);
  // low-rank spectral factors S0 (axis0) / S1 (axis1)
  {
    int t0 = MH * 2 * 128 * 128;
    buildS_kernel<<<(t0 + 255) / 256, 256, 0, stream>>>(kout0, kin0, S0r, S0i, t0);
    int t1 = MW * 2 * 128 * 128;
    buildS_kernel<<<(t1 + 255) / 256, 256, 0, stream>>>(kout1, kin1, S1r, S1i, t1);
  }
  // zero Y (for w-padding rows 17..19)
  {
    int n = (int)(2 * Y_PLANE);  // Yr,Yi contiguous
    zero_kernel<<<(n + 255) / 256, 256, 0, stream>>>(Yr, n);
  }

  fwdW_kernel<<<NB * NH, 256, 0, stream>>>(X, FwR, FwI, T1r, T1i);
  fwdH_kernel<<<NB * MW, 256, 0, stream>>>(T1r, T1i, FhR, FhI, Xfr, Xfi);
  {
    dim3 grid(MH * MW, NC / 16);
    spectral_kernel<<<grid, 64, 0, stream>>>(Xfr, Xfi, S0r, S0i, S1r, S1i, Yr, Yi);
  }
  invH_kernel<<<NB * MWP, 256, 0, stream>>>(Yr, Yi, GhR, GhI, T2r, T2i);
  invW_kernel<<<NB * NH, 256, 0, stream>>>(T2r, T2i, GwC, GwS, out);
}